// SoftDTW_90503550861914
// MI455X (gfx1250) — compile-verified
//
#include <hip/hip_runtime.h>

// Soft-DTW, B=128, T=512, gamma=0.1.
// Latency-bound DP -> tile-wavefront (16x16 tiles, 63 tile-diagonals, 2 barriers each)
// with an intra-half-wave (16-lane) systolic pipeline. Neighbor values flow on
// DPP16 row_shr:1 / row_shl:1 (1-cycle VALU lane shifts) instead of ds_bpermute,
// keeping the serial softmin recurrence off the LDS pipeline entirely.
// D-tiles produced on the WMMA pipe via rank-3 factorization:
//   (x-y)^2 = [x^2, x, 1] . [1, -2y, y^2]^T   ->  V_WMMA_F32_16X16X4_F32

#define GAMMA_F 0.1f
#define INVG_F 10.0f
#define BIGV 10000000.0f
#define TLEN 512
#define NBATCH 128
#define TILE 16
#define NT 32              // tiles per dimension (512/16)
#define NDIAG (2 * NT - 1) // 63 tile anti-diagonals

typedef __attribute__((ext_vector_type(2))) float v2f;
typedef __attribute__((ext_vector_type(8))) float v8f;

// ---- 16-lane shifts: DPP16 when available (VALU-latency), shfl fallback ----
#if __has_builtin(__builtin_amdgcn_update_dpp)
__device__ __forceinline__ float lane_shr1(float v) { // lane n <- lane n-1 (row of 16)
    return __int_as_float(__builtin_amdgcn_update_dpp(
        0, __float_as_int(v), 0x111 /*row_shr:1*/, 0xf, 0xf, true));
}
__device__ __forceinline__ float lane_shl1(float v) { // lane n <- lane n+1 (row of 16)
    return __int_as_float(__builtin_amdgcn_update_dpp(
        0, __float_as_int(v), 0x101 /*row_shl:1*/, 0xf, 0xf, true));
}
#else
__device__ __forceinline__ float lane_shr1(float v) { return __shfl_up(v, 1, 16); }
__device__ __forceinline__ float lane_shl1(float v) { return __shfl_down(v, 1, 16); }
#endif

__device__ __forceinline__ float softmin3(float a, float b, float c) {
    // -g*log(e^-a/g + e^-b/g + e^-c/g), min-shifted; base-2 HW trans ops,
    // constants pre-folded: K2 = (1/g)*log2(e), G2 = g*ln(2)
    const float K2 = INVG_F * 1.44269504088896341f;
    const float G2 = GAMMA_F * 0.69314718055994531f;
    float m = fminf(a, fminf(b, c));
    float s = __builtin_amdgcn_exp2f((m - a) * K2) +
              __builtin_amdgcn_exp2f((m - b) * K2) +
              __builtin_amdgcn_exp2f((m - c) * K2);
    return m - G2 * __builtin_amdgcn_logf(s);
}

__launch_bounds__(512, 1)
__global__ void sdtw_kernel(const float* __restrict__ x,
                            const float* __restrict__ y,
                            float* __restrict__ perBatch) {
    __shared__ float sx[TLEN], sy[TLEN];
    __shared__ float Hbuf[TLEN];          // R[row_last][c] per column c
    __shared__ float Vbuf[TLEN];          // R[r][col_last] per row r
    __shared__ float CBuf[2][NT];         // double-buffered corner hand-off
    __shared__ float sd[NT][TILE * TILE]; // D tiles for current diagonal

    const int b    = blockIdx.x;
    const int t    = threadIdx.x; // 0..511
    const int half = t >> 4;      // 0..31 : tile slot on the diagonal
    const int l    = t & 15;      // row within tile
    const int wave = t >> 5;      // 0..15
    const int lane = t & 31;

    sx[t]   = x[b * TLEN + t];
    sy[t]   = y[b * TLEN + t];
    Hbuf[t] = BIGV;               // R[-1][c] = BIG
    Vbuf[t] = BIGV;               // R[r][-1] = BIG
    if (t < 2 * NT) ((float*)CBuf)[t] = BIGV;
    __syncthreads();

    for (int td = 0; td < NDIAG; ++td) {
        const int tstart = (td > NT - 1) ? (td - (NT - 1)) : 0;
        const int tend   = (td < NT - 1) ? td : (NT - 1);
        const int n_t    = tend - tstart + 1;

        const int  ti    = tstart + half;
        const int  tj    = td - ti;
        const bool valid = (half < n_t);

        // ---------- Phase A: read borders, publish corner for right neighbor ----------
        float lb = BIGV, topb = BIGV, corner = BIGV;
        if (valid) {
            lb   = (tj == 0) ? BIGV : Vbuf[ti * TILE + l];       // R[i][tj*16-1]
            topb = (ti == 0) ? BIGV : Hbuf[tj * TILE + l];       // R[ti*16-1][tj*16+l]
            if (l == 0) {
                corner = (ti == 0) ? ((tj == 0) ? 0.0f : BIGV)
                                   : ((tj == 0) ? BIGV : CBuf[(td + 1) & 1][tj - 1]);
            }
            if (l == 15) {
                // top-right of this tile's top border == corner of (ti, tj+1)
                CBuf[td & 1][tj] = topb;
            }
        }

        // ---------- D tiles for this diagonal (matrix pipe) ----------
#if __has_builtin(__builtin_amdgcn_wmma_f32_16x16x4_f32)
        // Wave-uniform branches; EXEC all-ones inside as WMMA requires.
        #pragma unroll
        for (int h2 = 0; h2 < 2; ++h2) {
            const int k = wave * 2 + h2; // slot on diagonal; uniform per wave
            if (k < n_t) {
                const int   wti = tstart + k;
                const int   wtj = td - wti;
                const float xv  = sx[wti * TILE + (lane & 15)];
                const float yv  = sy[wtj * TILE + (lane & 15)];
                // A row m: [x^2, x, 1, 0]   (lanes 0-15: K0,K1 ; lanes 16-31: K2,K3)
                v2f A, Bv;
                A.x = (lane < 16) ? xv * xv : 1.0f;
                A.y = (lane < 16) ? xv : 0.0f;
                // B col n: [1, -2y, y^2, 0]^T (lanes 0-15: rows K0,K1 ; 16-31: K2,K3)
                Bv.x = (lane < 16) ? 1.0f : yv * yv;
                Bv.y = (lane < 16) ? -2.0f * yv : 0.0f;
                v8f C = {0.f, 0.f, 0.f, 0.f, 0.f, 0.f, 0.f, 0.f};
                C = __builtin_amdgcn_wmma_f32_16x16x4_f32(false, A, false, Bv,
                                                          (short)0, C, false, false);
                // C/D layout: lane n<16 -> rows 0..7 of col n; lane n+16 -> rows 8..15
                const int col   = lane & 15;
                const int rbase = (lane < 16) ? 0 : 8;
                #pragma unroll
                for (int r = 0; r < 8; ++r)
                    sd[k][(rbase + r) * TILE + col] = C[r];
            }
        }
#else
        if (valid) {
            const float xv = sx[ti * TILE + l];
            #pragma unroll
            for (int j = 0; j < TILE; ++j) {
                float d = xv - sy[tj * TILE + j];
                sd[half][l * TILE + j] = d * d;
            }
        }
#endif
        __syncthreads();

        // ---------- Phase B: 16-lane systolic DP, branchless, register-resident ----------
        if (valid) {
            const float lbm1 = lane_shr1(lb);   // lb[l-1]
            const int   hbase = tj * TILE;
            float tshift = topb;                // lane 0 holds topb[s] at step s
            float tPrev  = topb;                // lane 0: topb[s-1]
            float cur = 0.0f, upPrev = 0.0f;
            #pragma unroll
            for (int s = 0; s < 2 * TILE - 1; ++s) {
                const float fromAbove = lane_shr1(cur); // lane l-1's last value
                const int   j      = s - l;
                const bool  active = (j >= 0) && (j < TILE);
                const bool  l0     = (l == 0);
                const bool  j0     = (j == 0);
                const int   jj     = active ? j : 0;

                const float dval = sd[half][l * TILE + jj]; // off-chain LDS read
                const float up   = l0 ? tshift : fromAbove;              // R[i-1][j]
                const float dg   = j0 ? (l0 ? corner : lbm1)
                                      : (l0 ? tPrev  : upPrev);          // R[i-1][j-1]
                const float lf   = j0 ? lb : cur;                        // R[i][j-1]
                const float v    = dval + softmin3(up, lf, dg);
                cur = active ? v : cur;

                if (s >= TILE - 1) { // border exports exist only in the last 16 copies
                    if (l == 15 && active) Hbuf[hbase + j] = cur;      // bottom row
                    if (active && j == 15) Vbuf[ti * TILE + l] = cur;  // right col
                }
                upPrev = fromAbove;
                tPrev  = tshift;
                tshift = lane_shl1(tshift);
            }
            if (ti == NT - 1 && tj == NT - 1 && l == 15)
                perBatch[b] = cur; // R[511][511]
        }
        __syncthreads();
    }
}

__global__ void sdtw_reduce(const float* __restrict__ perBatch,
                            float* __restrict__ out) {
    __shared__ float ssum[NBATCH];
    const int t = threadIdx.x;
    ssum[t]     = perBatch[t];
    __syncthreads();
    for (int off = NBATCH / 2; off > 0; off >>= 1) {
        if (t < off) ssum[t] += ssum[t + off];
        __syncthreads();
    }
    if (t == 0) out[0] = ssum[0] * (1.0f / (float)NBATCH);
}

extern "C" void kernel_launch(void* const* d_in, const int* in_sizes, int n_in,
                              void* d_out, int out_size, void* d_ws, size_t ws_size,
                              hipStream_t stream) {
    (void)in_sizes; (void)n_in; (void)out_size; (void)ws_size;
    const float* x        = (const float*)d_in[0];
    const float* y        = (const float*)d_in[1];
    float*       perBatch = (float*)d_ws;

    sdtw_kernel<<<NBATCH, 512, 0, stream>>>(x, y, perBatch);
    sdtw_reduce<<<1, NBATCH, 0, stream>>>(perBatch, (float*)d_out);
}